// BaseXRayVolumeRenderer_27066883899444
// MI455X (gfx1250) — compile-verified
//
#include <hip/hip_runtime.h>
#include <math.h>

#define IMG_H 256
#define IMG_W 256
#define N_PTS 192
#define EPS 1e-8f

typedef float v2f __attribute__((ext_vector_type(2)));
typedef float v8f __attribute__((ext_vector_type(8)));

// ---------------------------------------------------------------------------
// Kernel 1: one thread per pixel; march 192 samples, masked trilinear taps.
// Volume (8MB) is L2-resident; 2048 wave32 waves hide gather latency.
// ---------------------------------------------------------------------------
__global__ __launch_bounds__(256) void xray_render(
    const float* __restrict__ vol, const float* __restrict__ Rm,
    const float* __restrict__ Tv, float* __restrict__ gray)
{
  const int idx = blockIdx.x * blockDim.x + threadIdx.x;
  const int w = idx & (IMG_W - 1);
  const int h = idx >> 8;

  // xs/ys = linspace(1,-1,256)
  const float xg = fmaf((float)w, (float)(-2.0 / 255.0), 1.0f);
  const float yg = fmaf((float)h, (float)(-2.0 / 255.0), 1.0f);

  const float r00 = Rm[0], r01 = Rm[1], r02 = Rm[2];
  const float r10 = Rm[3], r11 = Rm[4], r12 = Rm[5];
  const float r20 = Rm[6], r21 = Rm[7], r22 = Rm[8];
  const float t0 = Tv[0], t1 = Tv[1], t2 = Tv[2];

  // local = R*(cam - T), cam = depth*(xg/4, yg/4, 1)  ->  depth*a + o
  const float dxq = xg * 0.25f, dyq = yg * 0.25f;
  const float ax = r00 * dxq + r01 * dyq + r02;
  const float ay = r10 * dxq + r11 * dyq + r12;
  const float az = r20 * dxq + r21 * dyq + r22;
  const float ox = -(r00 * t0 + r01 * t1 + r02 * t2);
  const float oy = -(r10 * t0 + r11 * t1 + r12 * t2);
  const float oz = -(r20 * t0 + r21 * t1 + r22 * t2);

  float trans = 1.0f;  // running prod(1 - dens); (1+1e-10-d)==(1-d) in f32
  float rgb = 0.0f;

  for (int p = 0; p < N_PTS; ++p) {
    const float depth = fmaf((float)p, (float)(6.0 / 191.0), 3.0f);
    const float lx = fmaf(depth, ax, ox);
    const float ly = fmaf(depth, ay, oy);
    const float lz = fmaf(depth, az, oz);
    const float fx = ((lx + 1.0f) * 0.5f) * 127.0f;
    const float fy = ((ly + 1.0f) * 0.5f) * 127.0f;
    const float fz = ((lz + 1.0f) * 0.5f) * 127.0f;

    // wave-coherent skip: outside -> dens=0, factor (1-0)=1, contributes 0
    if (fx > -1.0f && fx < 128.0f && fy > -1.0f && fy < 128.0f &&
        fz > -1.0f && fz < 128.0f) {
      const float x0 = floorf(fx), y0 = floorf(fy), z0 = floorf(fz);
      const float wx = fx - x0, wy = fy - y0, wz = fz - z0;
      // per-corner validity masks folded into the weights
      const float wx0 = (1.0f - wx) * ((x0 >= 0.0f) ? 1.0f : 0.0f);
      const float wx1 = wx * ((x0 <= 126.0f) ? 1.0f : 0.0f);
      const float wy0 = (1.0f - wy) * ((y0 >= 0.0f) ? 1.0f : 0.0f);
      const float wy1 = wy * ((y0 <= 126.0f) ? 1.0f : 0.0f);
      const float wz0 = (1.0f - wz) * ((z0 >= 0.0f) ? 1.0f : 0.0f);
      const float wz1 = wz * ((z0 <= 126.0f) ? 1.0f : 0.0f);
      const int ix0 = (int)fminf(fmaxf(x0, 0.0f), 127.0f);
      const int iy0 = (int)fminf(fmaxf(y0, 0.0f), 127.0f);
      const int iz0 = (int)fminf(fmaxf(z0, 0.0f), 127.0f);
      const int ix1 = (int)fminf(fmaxf(x0 + 1.0f, 0.0f), 127.0f);
      const int iy1 = (int)fminf(fmaxf(y0 + 1.0f, 0.0f), 127.0f);
      const int iz1 = (int)fminf(fmaxf(z0 + 1.0f, 0.0f), 127.0f);

      const int b00 = (iz0 * 128 + iy0) * 128;
      const int b01 = (iz0 * 128 + iy1) * 128;
      const int b10 = (iz1 * 128 + iy0) * 128;
      const int b11 = (iz1 * 128 + iy1) * 128;
      const float wzy00 = wz0 * wy0, wzy01 = wz0 * wy1;
      const float wzy10 = wz1 * wy0, wzy11 = wz1 * wy1;

      float v = 0.0f, ds = 0.0f, wt;
      wt = wzy00 * wx0; v = fmaf(wt, vol[b00 + ix0], v); ds += wt;
      wt = wzy00 * wx1; v = fmaf(wt, vol[b00 + ix1], v); ds += wt;
      wt = wzy01 * wx0; v = fmaf(wt, vol[b01 + ix0], v); ds += wt;
      wt = wzy01 * wx1; v = fmaf(wt, vol[b01 + ix1], v); ds += wt;
      wt = wzy10 * wx0; v = fmaf(wt, vol[b10 + ix0], v); ds += wt;
      wt = wzy10 * wx1; v = fmaf(wt, vol[b10 + ix1], v); ds += wt;
      wt = wzy11 * wx0; v = fmaf(wt, vol[b11 + ix0], v); ds += wt;
      wt = wzy11 * wx1; v = fmaf(wt, vol[b11 + ix1], v); ds += wt;

      const float dens = ds * (1.0f / (float)N_PTS);
      rgb = fmaf(dens * trans, v, rgb);   // weight = dens * absorption
      trans *= (1.0f - dens);
    }
  }
  const float opac = 1.0f - trans;
  gray[idx] = (rgb + rgb + rgb + opac) * 0.25f;  // channel mean of (r,r,r,a)
}

// ---------------------------------------------------------------------------
// Kernel 2: single block, 8 waves. WMMA f32 16x16x4 with a ones B-matrix
// computes 16 f32 row-sums per instruction (sum and sum-of-squares);
// min/max ride along in VALU. Cross-lane via shfl_xor (wave32).
// ---------------------------------------------------------------------------
__global__ __launch_bounds__(256) void xray_reduce(const float* __restrict__ g,
                                                   float* __restrict__ scal)
{
  const int tid = threadIdx.x;
  const int wave = tid >> 5;
  const int lane = tid & 31;
  // A-matrix 16x4 layout: lanes 0-15 hold K=0,1; lanes 16-31 hold K=2,3
  const int laneOff = (lane & 15) * 4 + (lane >> 4) * 2;
  const int base = wave * 8192 + laneOff;

  v8f accS = {0.f, 0.f, 0.f, 0.f, 0.f, 0.f, 0.f, 0.f};
  v8f accQ = {0.f, 0.f, 0.f, 0.f, 0.f, 0.f, 0.f, 0.f};
  const v2f ones = {1.0f, 1.0f};
  float vmin = 3.402823466e38f, vmax = -3.402823466e38f;

  for (int it = 0; it < 128; ++it) {
    const float* p = g + base + it * 64;
    v2f a; a.x = p[0]; a.y = p[1];
    vmin = fminf(vmin, fminf(a.x, a.y));
    vmax = fmaxf(vmax, fmaxf(a.x, a.y));
    v2f a2 = a * a;
    accS = __builtin_amdgcn_wmma_f32_16x16x4_f32(false, a,  false, ones,
                                                 (short)0, accS, false, false);
    accQ = __builtin_amdgcn_wmma_f32_16x16x4_f32(false, a2, false, ones,
                                                 (short)0, accQ, false, false);
  }
  // C/D: VGPR r holds rows r (lanes 0-15) and r+8 (lanes 16-31); all N equal.
  float s = 0.f, q = 0.f;
#pragma unroll
  for (int i = 0; i < 8; ++i) { s += accS[i]; q += accQ[i]; }
  s += __shfl_xor(s, 16, 32);
  q += __shfl_xor(q, 16, 32);
#pragma unroll
  for (int m = 16; m >= 1; m >>= 1) {
    vmin = fminf(vmin, __shfl_xor(vmin, m, 32));
    vmax = fmaxf(vmax, __shfl_xor(vmax, m, 32));
  }

  __shared__ float sh[4][8];
  if (lane == 0) { sh[0][wave] = s; sh[1][wave] = q;
                   sh[2][wave] = vmin; sh[3][wave] = vmax; }
  __syncthreads();
  if (tid == 0) {
    double S = 0.0, Q = 0.0;
    float mn = sh[2][0], mx = sh[3][0];
    for (int i = 0; i < 8; ++i) {
      S += (double)sh[0][i]; Q += (double)sh[1][i];
      mn = fminf(mn, sh[2][i]); mx = fmaxf(mx, sh[3][i]);
    }
    const double n = (double)(IMG_H * IMG_W);
    const double mean = S / n;
    double var = (Q - S * S / n) / (n - 1.0);   // ddof=1
    if (var < 0.0) var = 0.0;
    const float sden = (float)sqrt(var) + EPS;
    const float fmean = (float)mean;
    const float stdmin = (mn - fmean) / sden;
    const float stdmax = (mx - fmean) / sden;
    scal[0] = fmean;
    scal[1] = sden;
    scal[2] = stdmin;
    scal[3] = (stdmax - stdmin) + EPS;
  }
}

// ---------------------------------------------------------------------------
// Kernel 3: in-place standardize + min-max normalize.
// ---------------------------------------------------------------------------
__global__ __launch_bounds__(256) void xray_finalize(float* __restrict__ out,
                                                     const float* __restrict__ scal)
{
  const int idx = blockIdx.x * blockDim.x + threadIdx.x;
  const float mean = scal[0], sden = scal[1], stdmin = scal[2], denom = scal[3];
  const float sd = (out[idx] - mean) / sden;
  out[idx] = (sd - stdmin + EPS) / denom;
}

extern "C" void kernel_launch(void* const* d_in, const int* in_sizes, int n_in,
                              void* d_out, int out_size, void* d_ws, size_t ws_size,
                              hipStream_t stream)
{
  (void)in_sizes; (void)n_in; (void)out_size; (void)ws_size;
  const float* vol = (const float*)d_in[0];
  const float* Rm  = (const float*)d_in[1];
  const float* Tv  = (const float*)d_in[2];
  float* out  = (float*)d_out;
  float* scal = (float*)d_ws;   // 4 scalars

  xray_render  <<<(IMG_H * IMG_W) / 256, 256, 0, stream>>>(vol, Rm, Tv, out);
  xray_reduce  <<<1, 256, 0, stream>>>(out, scal);
  xray_finalize<<<(IMG_H * IMG_W) / 256, 256, 0, stream>>>(out, scal);
}